// BP_85538568667584
// MI455X (gfx1250) — compile-verified
//
#include <hip/hip_runtime.h>

#define TPB 256       // 8 wave32 waves per block
#define N_ROUNDS 10

// ---------- CDNA5 async global->LDS helpers (gfx1250, ASYNCcnt path) ----------
// Per-lane 16B copy: global memory -> LDS, tracked by ASYNCcnt.
__device__ __forceinline__ void async_ld_b128(unsigned int lds_addr, const void* gaddr) {
    asm volatile("global_load_async_to_lds_b128 %0, %1, off"
                 :: "v"(lds_addr), "v"(gaddr) : "memory");
}
__device__ __forceinline__ void wait_asynccnt0() {
    asm volatile("s_wait_asynccnt 0x0" ::: "memory");
}
// Generic (flat) addresses in the LDS aperture carry the LDS byte address in
// their low 32 bits (ISA 10.2).
__device__ __forceinline__ unsigned int lds_lo32(const void* p) {
    return (unsigned int)(size_t)p;
}

__global__ void k_zero(float* __restrict__ p, int n) {
    int i = blockIdx.x * blockDim.x + threadIdx.x;
    if (i < n) p[i] = 0.0f;
}

// c2l_msg[scat[r]] += c2l[rep[r]]  for r in [0, R1).
// Index quads are staged global->LDS asynchronously; the random gathers and
// non-returning L2 float atomics own the LOADcnt/STOREcnt paths.
__global__ void k_gather_scatter(const float* __restrict__ src,
                                 const int4* __restrict__ rep,
                                 const int4* __restrict__ scat,
                                 float* __restrict__ dst, int nq) {
    __shared__ int4 srep[TPB];
    __shared__ int4 sscat[TPB];
    int i = blockIdx.x * blockDim.x + threadIdx.x;
    if (i < nq) {
        async_ld_b128(lds_lo32(&srep[threadIdx.x]),  rep  + i);
        async_ld_b128(lds_lo32(&sscat[threadIdx.x]), scat + i);
        // warm L2 ahead on the sequential index streams (global_prefetch_b8)
        __builtin_prefetch(rep + i + 2048, 0, 1);
        __builtin_prefetch(scat + i + 2048, 0, 1);
    }
    wait_asynccnt0();
    if (i >= nq) return;
    int4 r = srep[threadIdx.x];
    int4 s = sscat[threadIdx.x];
    float vx = src[r.x], vy = src[r.y], vz = src[r.z], vw = src[r.w];
    atomicAdd(dst + s.x, vx);
    atomicAdd(dst + s.y, vy);
    atomicAdd(dst + s.z, vz);
    atomicAdd(dst + s.w, vw);
}

// scalar tail in case R1 % 4 != 0
__global__ void k_gather_scatter_tail(const float* __restrict__ src,
                                      const int* __restrict__ rep,
                                      const int* __restrict__ scat,
                                      float* __restrict__ dst,
                                      int start, int n) {
    int i = start + blockIdx.x * blockDim.x + threadIdx.x;
    if (i >= n) return;
    atomicAdd(dst + scat[i], src[rep[i]]);
}

// l2c[2i], l2c[2i+1] = pair - logsumexp(pair)   (polarity normalization)
__global__ void k_pair_norm(const float2* __restrict__ c2v,
                            float2* __restrict__ l2c, int npairs) {
    int i = blockIdx.x * blockDim.x + threadIdx.x;
    if (i >= npairs) return;
    float2 p = c2v[i];
    float m = fmaxf(p.x, p.y);
    float lse = m + __logf(__expf(p.x - m) + __expf(p.y - m));
    float2 o; o.x = p.x - lse; o.y = p.y - lse;
    l2c[i] = o;
}

// Fused: aggr[m]=l2c[arep[2m]]+l2c[arep[2m+1]] ; c2l[e]=LSE(aggr[2e],aggr[2e+1]).
// Also re-zeroes c2l_msg[e] for the next round's atomics (fused k_zero).
__global__ void k_aggr_lse(const float* __restrict__ l2c,
                           const int4* __restrict__ arep,
                           float* __restrict__ c2l,
                           float* __restrict__ c2l_msg_zero, int E) {
    __shared__ int4 sidx[TPB];
    int e = blockIdx.x * blockDim.x + threadIdx.x;
    if (e < E) {
        async_ld_b128(lds_lo32(&sidx[threadIdx.x]), arep + e);
        __builtin_prefetch(arep + e + 2048, 0, 1);
    }
    wait_asynccnt0();
    if (e >= E) return;
    int4 r = sidx[threadIdx.x];
    float a = l2c[r.x] + l2c[r.y];
    float b = l2c[r.z] + l2c[r.w];
    float m = fmaxf(a, b);
    c2l[e] = m + __logf(__expf(a - m) + __expf(b - m));
    c2l_msg_zero[e] = 0.0f;   // re-arm accumulator for next round
}

// l_logit[sidx[e]] += c2l[e], vectorized 4-wide
__global__ void k_final_scatter(const float* __restrict__ c2l,
                                const int* __restrict__ sidx,
                                float* __restrict__ llogit, int E) {
    int base = (blockIdx.x * blockDim.x + threadIdx.x) * 4;
    if (base + 3 < E) {
        int4   s = *(const int4*)(sidx + base);
        float4 v = *(const float4*)(c2l + base);
        atomicAdd(llogit + s.x, v.x);
        atomicAdd(llogit + s.y, v.y);
        atomicAdd(llogit + s.z, v.z);
        atomicAdd(llogit + s.w, v.w);
    } else {
        for (int e = base; e < E; ++e)
            atomicAdd(llogit + sidx[e], c2l[e]);
    }
}

__global__ void k_softmax(const float2* __restrict__ llogit,
                          float2* __restrict__ out, int V) {
    int v = blockIdx.x * blockDim.x + threadIdx.x;
    if (v >= V) return;
    float2 p = llogit[v];
    float m  = fmaxf(p.x, p.y);
    float ea = __expf(p.x - m), eb = __expf(p.y - m);
    float inv = 1.0f / (ea + eb);
    float2 o; o.x = ea * inv; o.y = eb * inv;
    out[v] = o;
}

extern "C" void kernel_launch(void* const* d_in, const int* in_sizes, int n_in,
                              void* d_out, int out_size, void* d_ws, size_t ws_size,
                              hipStream_t stream) {
    const float* c2l_init = (const float*)d_in[0];
    const int*   sign_idx = (const int*)d_in[1];
    const int*   rep_s    = (const int*)d_in[2];
    const int*   scat_s   = (const int*)d_in[3];
    const int4*  rep      = (const int4*)d_in[2];
    const int4*  scat     = (const int4*)d_in[3];
    const int4*  arep     = (const int4*)d_in[4];
    // d_in[5], d_in[6] are repeat(arange(...),2) scatter indices -> folded into
    // the fused kernels. d_in[7] (l_size) == out_size.

    const int E     = in_sizes[0];       // 504000
    const int R1    = in_sizes[2];       // 6,048,000
    const int lsize = out_size;          // 2V = 40000
    const int V     = lsize / 2;

    float* c2l     = (float*)d_ws;       // E
    float* c2l_msg = c2l + E;            // E
    float* l2c     = c2l_msg + E;        // E
    float* llog    = l2c + E;            // 2V

    hipMemcpyAsync(c2l, c2l_init, (size_t)E * sizeof(float),
                   hipMemcpyDeviceToDevice, stream);

    const int nq   = R1 >> 2;
    const int tail = R1 & 3;
    dim3 b(TPB);
    auto g = [](int n) { return dim3((unsigned)((n + TPB - 1) / TPB)); };

    k_zero<<<g(E), b, 0, stream>>>(c2l_msg, E);   // arm accumulator once
    for (int round = 0; round < N_ROUNDS; ++round) {
        k_gather_scatter<<<g(nq), b, 0, stream>>>(c2l, rep, scat, c2l_msg, nq);
        if (tail)
            k_gather_scatter_tail<<<1, TPB, 0, stream>>>(c2l, rep_s, scat_s,
                                                         c2l_msg, nq * 4, R1);
        k_pair_norm<<<g(E / 2), b, 0, stream>>>((const float2*)c2l_msg,
                                                (float2*)l2c, E / 2);
        k_aggr_lse<<<g(E), b, 0, stream>>>(l2c, arep, c2l, c2l_msg, E);
    }

    k_zero<<<g(lsize), b, 0, stream>>>(llog, lsize);
    k_final_scatter<<<g((E + 3) / 4), b, 0, stream>>>(c2l, sign_idx, llog, E);
    k_softmax<<<g(V), b, 0, stream>>>((const float2*)llog, (float2*)d_out, V);
}